// EfficientGlobalPointer_15238543966613
// MI455X (gfx1250) — compile-verified
//
#include <hip/hip_runtime.h>

#define B_    8
#define S_    512
#define HID_  1024
#define D_    64
#define H_    16
#define TWOD_ 128
#define TWOH_ 32

typedef float v8f __attribute__((ext_vector_type(8)));
typedef float v2f __attribute__((ext_vector_type(2)));

// ---------------------------------------------------------------------------
// Kernel 1: per block -> 16 rows of x = inputs@W1 + b1 (f32 WMMA, K=1024),
// then RoPE -> qw/kw, and bias = (x@W2 + b2)/2.
// Block: 128 threads = 4 waves; wave w owns N-tiles {2w, 2w+1} of the 128 cols.
// ---------------------------------------------------------------------------
__global__ __launch_bounds__(128) void k1_x_rope_bias(
    const float* __restrict__ inputs, const float* __restrict__ W1,
    const float* __restrict__ b1, const float* __restrict__ W2,
    const float* __restrict__ b2,
    float* __restrict__ qw, float* __restrict__ kw, float* __restrict__ biaso)
{
    __shared__ float As[16][36];     // 16x32 input chunk (padded)
    __shared__ float Ws[32][128];    // 32x128 W1 chunk
    __shared__ float xs[16][132];    // 16x128 x tile (padded)

    const int t    = threadIdx.x;
    const int lane = t & 31;
    const int wv   = t >> 5;            // 0..3
    const int half = lane >> 4;         // 0 or 1
    const int l15  = lane & 15;

    const int row0 = blockIdx.x * 16;   // flattened row in [0, B*S)
    const int b    = row0 / S_;
    const int s0   = row0 % S_;

    v8f acc0 = {}; v8f acc1 = {};
    const int nt0 = (2 * wv) * 16;
    const int nt1 = (2 * wv + 1) * 16;

    for (int k0 = 0; k0 < HID_; k0 += 32) {
        // stage 16x32 of inputs: one float4 per thread
        {
            int r  = t >> 3;
            int c4 = (t & 7) * 4;
            float4 v = *(const float4*)&inputs[(size_t)(row0 + r) * HID_ + k0 + c4];
            As[r][c4 + 0] = v.x; As[r][c4 + 1] = v.y;
            As[r][c4 + 2] = v.z; As[r][c4 + 3] = v.w;
        }
        // stage 32x128 of W1: 8 float4 per thread
        #pragma unroll
        for (int i = 0; i < 8; ++i) {
            int idx = t + i * 128;          // float4 index 0..1023
            int r   = idx >> 5;             // 0..31
            int c4  = (idx & 31) * 4;
            *(float4*)&Ws[r][c4] = *(const float4*)&W1[(size_t)(k0 + r) * TWOD_ + c4];
        }
        __syncthreads();

        #pragma unroll
        for (int kk = 0; kk < 32; kk += 4) {
            v2f a, f0, f1;
            a.x  = As[l15][kk + 2 * half + 0];
            a.y  = As[l15][kk + 2 * half + 1];
            f0.x = Ws[kk + 2 * half + 0][nt0 + l15];
            f0.y = Ws[kk + 2 * half + 1][nt0 + l15];
            f1.x = Ws[kk + 2 * half + 0][nt1 + l15];
            f1.y = Ws[kk + 2 * half + 1][nt1 + l15];
            acc0 = __builtin_amdgcn_wmma_f32_16x16x4_f32(false, a, false, f0,
                                                         (short)0, acc0, false, false);
            acc1 = __builtin_amdgcn_wmma_f32_16x16x4_f32(false, a, false, f1,
                                                         (short)0, acc1, false, false);
        }
        __syncthreads();
    }

    // spill accumulators (+b1) to LDS: C element r -> row r + 8*half, col = nt + l15
    #pragma unroll
    for (int r = 0; r < 8; ++r) {
        int rr = r + 8 * half;
        xs[rr][nt0 + l15] = acc0[r] + b1[nt0 + l15];
        xs[rr][nt1 + l15] = acc1[r] + b1[nt1 + l15];
    }
    __syncthreads();

    // RoPE -> qw / kw. thread handles row = t>>3, 16 consecutive cols.
    {
        int r = t >> 3;
        int s = s0 + r;
        int cbase = (t & 7) * 16;
        for (int c = cbase; c < cbase + 16; ++c) {
            int   d    = (c < D_) ? c : (c - D_);
            int   i    = d >> 1;
            float freq = powf(10000.0f, -(float)i * (1.0f / 32.0f));
            float ang  = (float)s * freq;
            float sp, cp;
            sincosf(ang, &sp, &cp);
            float xv  = xs[r][c];
            float rot = (d & 1) ? xs[r][c - 1] : -xs[r][c + 1];
            float o   = xv * cp + rot * sp;
            if (c < D_) qw[(size_t)(b * S_ + s) * D_ + d] = o;
            else        kw[(size_t)(b * S_ + s) * D_ + d] = o;
        }
    }

    // bias[b, j, s] = (x @ W2 + b2)/2   (16 rows x 32 outputs)
    for (int o = t; o < 16 * 32; o += 128) {
        int r = o >> 5;
        int j = o & 31;
        float sum = b2[j];
        for (int k = 0; k < TWOD_; ++k)
            sum += xs[r][k] * W2[k * TWOH_ + j];
        biaso[((size_t)b * TWOH_ + j) * S_ + (s0 + r)] = sum * 0.5f;
    }
}

// ---------------------------------------------------------------------------
// Kernel 2: one wave per 16x16 qk tile (f32 WMMA, K=64), broadcast across 16
// heads with row/col biases + mask + causal; non-temporal 128 MiB store.
// Block: 32x8 (8 waves sharing the LDS qw tile; each wave a distinct n-tile).
// ---------------------------------------------------------------------------
__global__ __launch_bounds__(256) void k2_logits(
    const float* __restrict__ qw, const float* __restrict__ kw,
    const float* __restrict__ biasi, const int* __restrict__ mask,
    float* __restrict__ out)
{
    __shared__ float qs[16][68];

    const int lane = threadIdx.x;                    // 0..31 (one wave per y)
    const int wv   = threadIdx.y;                    // 0..7
    const int half = lane >> 4;
    const int l15  = lane & 15;
    const int n0   = (blockIdx.x * 8 + wv) * 16;
    const int m0   = blockIdx.y * 16;
    const int b    = blockIdx.z;

    // stage qw tile 16x64 (shared by all 8 waves): one float4 per thread
    {
        int t  = wv * 32 + lane;
        int r  = t >> 4;
        int c4 = (t & 15) * 4;
        *(float4*)&qs[r][c4] = *(const float4*)&qw[(size_t)(b * S_ + m0 + r) * D_ + c4];
    }
    __syncthreads();

    v8f c = {};
    const float* kwrow = &kw[(size_t)(b * S_ + n0 + l15) * D_ + 2 * half];
    #pragma unroll
    for (int k = 0; k < D_; k += 4) {
        v2f a, f;
        a.x = qs[l15][k + 2 * half + 0];
        a.y = qs[l15][k + 2 * half + 1];
        float2 kv = *(const float2*)&kwrow[k];
        f.x = kv.x; f.y = kv.y;
        c = __builtin_amdgcn_wmma_f32_16x16x4_f32(false, a, false, f,
                                                  (short)0, c, false, false);
    }

    // per-lane row/col mask values
    float mrow[8];
    {
        const int* mp = &mask[b * S_ + m0 + half * 8];
        int4 a0 = *(const int4*)mp;
        int4 a1 = *(const int4*)(mp + 4);
        mrow[0] = (float)a0.x; mrow[1] = (float)a0.y;
        mrow[2] = (float)a0.z; mrow[3] = (float)a0.w;
        mrow[4] = (float)a1.x; mrow[5] = (float)a1.y;
        mrow[6] = (float)a1.z; mrow[7] = (float)a1.w;
    }
    const float mc   = (float)mask[b * S_ + n0 + l15];
    const float NEGV = -1e12f;
    const int   n    = n0 + l15;

    for (int h = 0; h < H_; ++h) {
        const float  bq  = biasi[((size_t)b * TWOH_ + h) * S_ + n];
        const float* bkp = &biasi[((size_t)b * TWOH_ + H_ + h) * S_ + m0 + half * 8];
        float4 bk0 = *(const float4*)bkp;
        float4 bk1 = *(const float4*)(bkp + 4);
        float  bk[8] = {bk0.x, bk0.y, bk0.z, bk0.w, bk1.x, bk1.y, bk1.z, bk1.w};
        float* op = &out[((size_t)(b * H_ + h) * S_ + m0 + half * 8) * S_ + n];
        #pragma unroll
        for (int r = 0; r < 8; ++r) {
            int   m = m0 + half * 8 + r;
            float v = c[r] * 0.125f + bq + bk[r];
            v = v * mrow[r] + NEGV * (1.0f - mrow[r]);
            v = v * mc      + NEGV * (1.0f - mc);
            if (m > n) v -= 1e12f;
            __builtin_nontemporal_store(v, op + (size_t)r * S_);
        }
    }
}

extern "C" void kernel_launch(void* const* d_in, const int* in_sizes, int n_in,
                              void* d_out, int out_size, void* d_ws, size_t ws_size,
                              hipStream_t stream) {
    const float* inputs = (const float*)d_in[0];
    const float* W1     = (const float*)d_in[1];
    const float* b1     = (const float*)d_in[2];
    const float* W2     = (const float*)d_in[3];
    const float* b2     = (const float*)d_in[4];
    const int*   mask   = (const int*)d_in[5];
    float*       out    = (float*)d_out;

    float* qw   = (float*)d_ws;
    float* kw   = qw + (size_t)B_ * S_ * D_;
    float* bias = kw + (size_t)B_ * S_ * D_;

    k1_x_rope_bias<<<dim3(B_ * S_ / 16), dim3(128), 0, stream>>>(
        inputs, W1, b1, W2, b2, qw, kw, bias);
    k2_logits<<<dim3(S_ / 16 / 8, S_ / 16, B_), dim3(32, 8), 0, stream>>>(
        qw, kw, bias, mask, out);
}